// MonotonicBahdanauAttention_1881195675869
// MI455X (gfx1250) — compile-verified
//
#include <hip/hip_runtime.h>

// MonotonicBahdanauAttention on MI455X (gfx1250, wave32).
// Memory-bound: 2 full passes over encoderHs (268 MB) ~= 23 us floor @ 23.3 TB/s.
// GEMM kept in fp32 via V_WMMA_F32_16X16X4_F32 (compute has huge headroom; avoid
// an extra conversion pass / precision loss of bf16).

typedef __attribute__((ext_vector_type(2))) float v2f;
typedef __attribute__((ext_vector_type(8))) float v8f;

#define B_ 32
#define T_ 4096
#define H_ 512
#define U_ 256
#define CLIP_LO 1e-10f

// ---------------- Kernel 0: dec_proj[b,u] = decoderHt[b,:] . Wa[:,u] ----------------
__global__ __launch_bounds__(256) void decproj_kernel(const float* __restrict__ dec,
                                                      const float* __restrict__ Wa,
                                                      float* __restrict__ dp) {
  const int b = blockIdx.x;
  const int u = threadIdx.x;
  float s = 0.f;
#pragma unroll 8
  for (int h = 0; h < H_; ++h)
    s += dec[b * H_ + h] * Wa[h * U_ + u];  // dec[..] is wave-uniform -> SMEM load
  dp[b * U_ + u] = s;
}

// ---------------- Kernel 1: fused enc_proj GEMM + tanh + Va dot -> score[B*T] -------
// Block = 8 waves, 128 rows of E = enc[B*T, H]; each wave: 16 rows x 256 cols.
// WMMA f32 16x16x4; K tiled by 32 through LDS.
#define ESTR 36  // padded LDS row stride (dwords), keeps 16B alignment, dodges bank conflicts

__global__ __launch_bounds__(256) void score_kernel(const float* __restrict__ enc,
                                                    const float* __restrict__ Uaw,
                                                    const float* __restrict__ Uab,
                                                    const float* __restrict__ Va,
                                                    const float* __restrict__ dp,
                                                    float* __restrict__ score) {
  __shared__ float Elds[128 * ESTR];  // 18 KB  : E tile 128 x 32
  __shared__ float Wlds[256 * ESTR];  // 36 KB  : W tile transposed [col][k]

  const int tid = threadIdx.x;
  const int wave = tid >> 5;
  const int lane = tid & 31;
  const int half = lane >> 4;   // 0: K+0/1, 1: K+2/3 (per ISA 32-bit A layout)
  const int l16 = lane & 15;
  const int R0 = blockIdx.x * 128;  // global row base; 128 | 4096 so one batch per block
  const int b = R0 >> 12;

  v8f acc[16];
#pragma unroll
  for (int n = 0; n < 16; ++n) acc[n] = {};

  const int wrow = wave * 16;
  const int aoff = (wrow + l16) * ESTR + half * 2;  // A-fragment LDS base (dwords)

  for (int kb = 0; kb < H_; kb += 32) {
    // Stage E tile: 128 rows x 32 K, coalesced float4 from global.
#pragma unroll
    for (int i = 0; i < 4; ++i) {
      const int idx4 = tid + i * 256;          // 1024 float4 total
      const int row = idx4 >> 3;               // 8 float4 per row
      const int q = idx4 & 7;
      const float4 v = *(const float4*)(enc + (size_t)(R0 + row) * H_ + kb + q * 4);
      *(float4*)(&Elds[row * ESTR + q * 4]) = v;
    }
    // Stage W tile transposed: rows kb..kb+31 of Ua_w, all 256 cols.
#pragma unroll
    for (int i = 0; i < 8; ++i) {
      const int idx4 = tid + i * 256;          // 2048 float4 total
      const int r = idx4 >> 6;                 // 64 float4 per K-row
      const int c = (idx4 & 63) * 4;
      const float4 v = *(const float4*)(Uaw + (size_t)(kb + r) * U_ + c);
      Wlds[(c + 0) * ESTR + r] = v.x;
      Wlds[(c + 1) * ESTR + r] = v.y;
      Wlds[(c + 2) * ESTR + r] = v.z;
      Wlds[(c + 3) * ESTR + r] = v.w;
    }
    __syncthreads();

#pragma unroll 2
    for (int k0 = 0; k0 < 32; k0 += 4) {
      // A frag: lane m -> {E[m][k0+2h], E[m][k0+1+2h]} : one ds_load_b64
      const v2f a = *(const v2f*)(&Elds[aoff + k0]);
#pragma unroll
      for (int n = 0; n < 16; ++n) {
        const v2f bf = *(const v2f*)(&Wlds[(n * 16 + l16) * ESTR + half * 2 + k0]);
        acc[n] = __builtin_amdgcn_wmma_f32_16x16x4_f32(
            /*neg_a=*/false, a, /*neg_b=*/false, bf,
            /*c_mod=*/(short)0, acc[n], /*reuse_a=*/false, /*reuse_b=*/false);
      }
    }
    __syncthreads();
  }

  // Epilogue: + dec_proj + bias, tanh, dot with Va, reduce over cols.
  float s[8];
#pragma unroll
  for (int r = 0; r < 8; ++r) s[r] = 0.f;
#pragma unroll
  for (int n = 0; n < 16; ++n) {
    const int col = n * 16 + l16;
    const float dpv = dp[b * U_ + col] + Uab[col];
    const float va = Va[col];
#pragma unroll
    for (int r = 0; r < 8; ++r) s[r] += tanhf(acc[n][r] + dpv) * va;
  }
  // C layout: VGPR r, lanes 0-15 -> row wrow+r, lanes 16-31 -> row wrow+8+r.
#pragma unroll
  for (int r = 0; r < 8; ++r) {
    float v = s[r];
    v += __shfl_xor(v, 1, 16);
    v += __shfl_xor(v, 2, 16);
    v += __shfl_xor(v, 4, 16);
    v += __shfl_xor(v, 8, 16);
    if (l16 == 0) score[R0 + wrow + half * 8 + r] = v;
  }
}

// ---------------- Kernel 2: monotonic attention scan (per batch) --------------------
__global__ __launch_bounds__(256) void mono_kernel(const float* __restrict__ score,
                                                   const float* __restrict__ prevAtt,
                                                   float* __restrict__ att) {
  __shared__ float sc[256];
  const int b = blockIdx.x;
  const int tid = threadIdx.x;
  const int base = b * T_ + tid * 16;  // 16 consecutive t per thread

  float p[16], cp[16], q[16];
  float tsum = 0.f;
#pragma unroll
  for (int i = 0; i < 16; ++i) {
    const float sv = score[base + i];
    const float pi = 1.f / (1.f + expf(-sv));
    p[i] = pi;
    const float li = logf(fminf(fmaxf(1.f - pi, CLIP_LO), 1.f));
    cp[i] = li;  // hold log(1-p) for now
    tsum += li;
  }
  // inclusive scan of per-thread sums (Hillis-Steele, deterministic)
  sc[tid] = tsum;
  __syncthreads();
  for (int off = 1; off < 256; off <<= 1) {
    const float v = (tid >= off) ? sc[tid - off] : 0.f;
    __syncthreads();
    sc[tid] += v;
    __syncthreads();
  }
  float run = sc[tid] - tsum;  // exclusive prefix across threads
  __syncthreads();

  // exclusive cumsum within thread -> cumprod_1mp; build q terms
  float qsum = 0.f;
#pragma unroll
  for (int i = 0; i < 16; ++i) {
    const float E = run;
    run += cp[i];
    const float c = expf(E);
    cp[i] = c;
    const float qi = prevAtt[base + i] / fminf(fmaxf(c, CLIP_LO), 1.f);
    q[i] = qi;
    qsum += qi;
  }
  sc[tid] = qsum;
  __syncthreads();
  for (int off = 1; off < 256; off <<= 1) {
    const float v = (tid >= off) ? sc[tid - off] : 0.f;
    __syncthreads();
    sc[tid] += v;
    __syncthreads();
  }
  float qrun = sc[tid] - qsum;  // exclusive prefix of q across threads
#pragma unroll
  for (int i = 0; i < 16; ++i) {
    qrun += q[i];  // inclusive cumsum of q
    att[base + i] = p[i] * cp[i] * qrun;
  }
}

// ---------------- Kernel 3: partial context over T chunks ---------------------------
#define NCH 64
#define TC (T_ / NCH)
__global__ __launch_bounds__(256) void ctxpart_kernel(const float* __restrict__ enc,
                                                      const float* __restrict__ att,
                                                      float* __restrict__ part) {
  const int b = blockIdx.x / NCH;
  const int ch = blockIdx.x % NCH;
  const int tid = threadIdx.x;
  const float* e = enc + ((size_t)b * T_ + (size_t)ch * TC) * H_;
  const float* a = att + b * T_ + ch * TC;
  float a0 = 0.f, a1 = 0.f;
#pragma unroll 4
  for (int t = 0; t < TC; ++t) {
    const float w = a[t];  // wave-uniform
    a0 += w * e[(size_t)t * H_ + tid];
    a1 += w * e[(size_t)t * H_ + tid + 256];
  }
  part[((size_t)ch * B_ + b) * H_ + tid] = a0;
  part[((size_t)ch * B_ + b) * H_ + tid + 256] = a1;
}

// ---------------- Kernel 4: deterministic reduction of partials ---------------------
__global__ __launch_bounds__(256) void ctxreduce_kernel(const float* __restrict__ part,
                                                        float* __restrict__ out) {
  const int idx = blockIdx.x * 256 + threadIdx.x;  // 0..B_*H_-1
  float s = 0.f;
#pragma unroll 8
  for (int ch = 0; ch < NCH; ++ch) s += part[(size_t)ch * (B_ * H_) + idx];
  out[idx] = s;
}

extern "C" void kernel_launch(void* const* d_in, const int* in_sizes, int n_in,
                              void* d_out, int out_size, void* d_ws, size_t ws_size,
                              hipStream_t stream) {
  const float* enc  = (const float*)d_in[0];  // [B,T,H]
  const float* dec  = (const float*)d_in[1];  // [B,H]
  const float* prev = (const float*)d_in[2];  // [B,T]
  const float* Wa   = (const float*)d_in[3];  // [H,U]
  const float* Uaw  = (const float*)d_in[4];  // [H,U]
  const float* Uab  = (const float*)d_in[5];  // [U]
  const float* Va   = (const float*)d_in[6];  // [U,1]
  float* out = (float*)d_out;                 // [B,H]

  float* ws    = (float*)d_ws;
  float* dp    = ws;                  // 32*256            = 8192 floats
  float* score = dp + B_ * U_;        // 32*4096           = 131072
  float* att   = score + B_ * T_;     // 32*4096           = 131072
  float* part  = att + B_ * T_;       // 64*32*512         = 1048576   (~5.1 MB total)

  decproj_kernel<<<B_, 256, 0, stream>>>(dec, Wa, dp);
  score_kernel<<<(B_ * T_) / 128, 256, 0, stream>>>(enc, Uaw, Uab, Va, dp, score);
  mono_kernel<<<B_, 256, 0, stream>>>(score, prev, att);
  ctxpart_kernel<<<B_ * NCH, 256, 0, stream>>>(enc, att, part);
  ctxreduce_kernel<<<(B_ * H_) / 256, 256, 0, stream>>>(part, out);
}